// db_store_32796370272788
// MI455X (gfx1250) — compile-verified
//
#include <hip/hip_runtime.h>
#include <hip/hip_bf16.h>
#include <stdint.h>

typedef __attribute__((ext_vector_type(2))) float v2f;
typedef __attribute__((ext_vector_type(8))) float v8f;

#define TILE_J 128              // j-rows staged in LDS per iteration
#define WAVES_PER_BLK 4         // one 16-row i-tile per wave
#define THREADS (WAVES_PER_BLK * 32)

// ---------------------------------------------------------------------------
// Prep: normalize db rows into a K=12-padded B matrix, compute per-patch
// thresholds t_i = 0.9*||p_i||, and copy db verbatim into the output head.
// ---------------------------------------------------------------------------
__global__ void prep_kernel(const float* __restrict__ pf,
                            const float* __restrict__ db,
                            float* __restrict__ dbn12,
                            float* __restrict__ thr,
                            float* __restrict__ out_db,
                            int n, int m, int m_pad) {
  int i = blockIdx.x * blockDim.x + threadIdx.x;
  if (i < m) {
    float v[9]; float s = 0.f;
#pragma unroll
    for (int k = 0; k < 9; ++k) { v[k] = db[i * 9 + k]; s += v[k] * v[k]; }
    float inv = 1.0f / sqrtf(s);
#pragma unroll
    for (int k = 0; k < 9; ++k) {
      dbn12[i * 12 + k] = v[k] * inv;
      out_db[i * 9 + k] = v[k];
    }
    dbn12[i * 12 + 9]  = 0.f;
    dbn12[i * 12 + 10] = 0.f;
    dbn12[i * 12 + 11] = 0.f;
  } else if (i < m_pad) {       // zero pad rows: dot==0 < thr, never trigger
#pragma unroll
    for (int k = 0; k < 12; ++k) dbn12[i * 12 + k] = 0.f;
  }
  if (i < n) {
    float s = 0.f;
#pragma unroll
    for (int k = 0; k < 9; ++k) { float x = pf[i * 9 + k]; s += x * x; }
    thr[i] = 0.9f * sqrtf(s);
  }
}

// ---------------------------------------------------------------------------
// Main: per-wave 16-row i-tile; loop over all j in LDS-staged 128-row tiles
// (async double-buffered); 3x V_WMMA_F32_16X16X4_F32 per 16x16 sub-tile
// (K padded 9->12); elementwise running max; cross-lane rowmax; write tail.
// ---------------------------------------------------------------------------
__global__ __launch_bounds__(THREADS) void cosfilter_kernel(
    const float* __restrict__ pf, const float* __restrict__ dbn12,
    const float* __restrict__ thr, float* __restrict__ out,
    int n, int m, int m_pad) {
  __shared__ __align__(16) float lds[2][TILE_J * 12];

  const int lane   = threadIdx.x & 31;
  const int wave   = threadIdx.x >> 5;
  const int tile_i = blockIdx.x * WAVES_PER_BLK + wave;
  const int i0     = tile_i * 16;
  const int arow   = i0 + (lane & 15);
  const int kb     = (lane < 16) ? 0 : 2;     // A/B half-wave K sub-offset

  // ---- A fragments: 16x12 (K chunks of 4), ISA 16x4 f32 A layout ----------
  v2f a[3];
#pragma unroll
  for (int kc = 0; kc < 3; ++kc) {
    int e0 = kc * 4 + kb;
    float x = (arow < n && e0     < 9) ? pf[arow * 9 + e0]     : 0.f;
    float y = (arow < n && e0 + 1 < 9) ? pf[arow * 9 + e0 + 1] : 0.f;
    a[kc].x = x; a[kc].y = y;
  }

  float rmax[8];
#pragma unroll
  for (int v = 0; v < 8; ++v) rmax[v] = -3.4e38f;

  const int NIT = m_pad / TILE_J;

  // ---- async prefetch of one 128-row B tile (6 KB): 3 x b128 per thread ---
  auto prefetch = [&](int stage, int j0) {
    const char* gbase = (const char*)dbn12 + (size_t)j0 * 48;  // 12 floats/row
    uint32_t lbase = (uint32_t)(uintptr_t)(&lds[stage][0]);    // LDS offset
#pragma unroll
    for (int r = 0; r < 3; ++r) {
      int chunk = r * THREADS + (int)threadIdx.x;              // 0..383
      uint32_t laddr = lbase + chunk * 16;
      const void* g  = gbase + chunk * 16;
      asm volatile("global_load_async_to_lds_b128 %0, %1, off"
                   :: "v"(laddr), "v"(g) : "memory");
    }
  };

  auto compute = [&](int stage) {
#pragma unroll
    for (int s = 0; s < 8; ++s) {
      // B layout mirror of A: lane L holds column N=L of rows K=kb,kb+1
      const float* bp = &lds[stage][(s * 16 + (lane & 15)) * 12 + kb];
      v2f b0 = *(const v2f*)(bp + 0);
      v2f b1 = *(const v2f*)(bp + 4);
      v2f b2 = *(const v2f*)(bp + 8);
      v8f c = {0.f, 0.f, 0.f, 0.f, 0.f, 0.f, 0.f, 0.f};
      c = __builtin_amdgcn_wmma_f32_16x16x4_f32(false, a[0], false, b0,
                                                (short)0, c, false, false);
      c = __builtin_amdgcn_wmma_f32_16x16x4_f32(false, a[1], false, b1,
                                                (short)0, c, false, false);
      c = __builtin_amdgcn_wmma_f32_16x16x4_f32(false, a[2], false, b2,
                                                (short)0, c, false, false);
#pragma unroll
      for (int v = 0; v < 8; ++v) rmax[v] = fmaxf(rmax[v], c[v]);
    }
  };

  prefetch(0, 0);
  for (int it = 0; it < NIT; ++it) {
    if (it + 1 < NIT) {
      prefetch((it + 1) & 1, (it + 1) * TILE_J);
      asm volatile("s_wait_asynccnt 3" ::: "memory");  // current tile landed
    } else {
      asm volatile("s_wait_asynccnt 0" ::: "memory");
    }
    __syncthreads();
    compute(it & 1);
    __syncthreads();   // all waves done reading before tile gets overwritten
  }

  // ---- rowmax across the 16 lanes of each half (xor stays inside half) ----
#pragma unroll
  for (int off = 8; off >= 1; off >>= 1)
#pragma unroll
    for (int v = 0; v < 8; ++v)
      rmax[v] = fmaxf(rmax[v], __shfl_xor(rmax[v], off, 32));

  // ---- write filtered patches: row v (lanes 0-15) / v+8 (lanes 16-31) -----
  const int e = lane & 15;
  const int halfsel = (lane < 16) ? 0 : 8;
#pragma unroll
  for (int v = 0; v < 8; ++v) {
    int r = i0 + v + halfsel;
    if (r < n && e < 9) {
      bool keep = rmax[v] < thr[r];
      float pv = pf[r * 9 + e];
      out[(size_t)(m + r) * 9 + e] = keep ? pv : 0.f;
    }
  }
}

// ---------------------------------------------------------------------------
extern "C" void kernel_launch(void* const* d_in, const int* in_sizes, int n_in,
                              void* d_out, int out_size, void* d_ws, size_t ws_size,
                              hipStream_t stream) {
  const float* pf = (const float*)d_in[0];   // patches, n x 9 row-major
  const float* db = (const float*)d_in[1];   // db,      m x 9 row-major
  float* out = (float*)d_out;                // (m+n) x 9

  int n = in_sizes[0] / 9;
  int m = in_sizes[1] / 9;
  int m_pad = ((m + TILE_J - 1) / TILE_J) * TILE_J;

  float* dbn12 = (float*)d_ws;                       // m_pad x 12 normalized
  float* thr   = dbn12 + (size_t)m_pad * 12;         // n thresholds

  int prep_n = (n > m_pad) ? n : m_pad;
  prep_kernel<<<(prep_n + 255) / 256, 256, 0, stream>>>(pf, db, dbn12, thr,
                                                        out, n, m, m_pad);

  int tiles  = (n + 15) / 16;                                    // 1024
  int blocks = (tiles + WAVES_PER_BLK - 1) / WAVES_PER_BLK;      // 256
  cosfilter_kernel<<<blocks, THREADS, 0, stream>>>(pf, dbn12, thr, out,
                                                   n, m, m_pad);
}